// SingleFullBandGAT_8899172237584
// MI455X (gfx1250) — compile-verified
//
#include <hip/hip_runtime.h>
#include <math.h>

typedef float v2f __attribute__((ext_vector_type(2)));
typedef float v8f __attribute__((ext_vector_type(8)));

#define NEG_SLOPE 0.2f
#define BN_EPS_C 1e-5f
// ordered-uint encoding of -inf: f2ord(0xFF800000) = ~0xFF800000
#define ORD_NEG_INF 0x007FFFFFu

__device__ __forceinline__ unsigned f2ord(float f) {
    unsigned u = __float_as_uint(f);
    return (u & 0x80000000u) ? ~u : (u | 0x80000000u);
}
__device__ __forceinline__ float ord2f(unsigned u) {
    return (u & 0x80000000u) ? __uint_as_float(u ^ 0x80000000u)
                             : __uint_as_float(~u);
}

// ---------------------------------------------------------------------------
// init: zero aggregation buffer, BN stats, denom, pool/cnt; emax = -inf
// ---------------------------------------------------------------------------
__global__ void __launch_bounds__(256)
init_layer(float* __restrict__ agg, unsigned* __restrict__ emax,
           float* __restrict__ denom, float* __restrict__ stats,
           float* __restrict__ pool, float* __restrict__ cnt, int N, int G) {
    long long i = (long long)blockIdx.x * 256 + threadIdx.x;
    long long n64 = (long long)N * 64;
    if (i < n64) agg[i] = 0.0f;
    if (i < N) { emax[i] = ORD_NEG_INF; denom[i] = 0.0f; }
    if (i < 256) stats[i] = 0.0f;
    if (i < (long long)G * 64) pool[i] = 0.0f;
    if (i < G) cnt[i] = 0.0f;
}

// ---------------------------------------------------------------------------
// GEMM1: H[N,64] = X[N,128] @ W[128,64]   (fp32 WMMA 16x16x4)
// one 16x64 output tile per wave; W staged in LDS transposed into K-pairs so
// each B fragment is one aligned ds_load_b64 into an even VGPR pair.
// ---------------------------------------------------------------------------
__global__ void __launch_bounds__(256)
gemm1_wmma(const float* __restrict__ X, const float* __restrict__ W,
           float* __restrict__ H, int nrows) {
    __shared__ v2f Ws[64 * 64];          // [kpair][col] = {W[2kp][col], W[2kp+1][col]}
    for (int j = threadIdx.x; j < 64 * 64; j += 256) {
        int kp = j >> 6;
        int col = j & 63;
        v2f w;
        w.x = W[(2 * kp) * 64 + col];
        w.y = W[(2 * kp + 1) * 64 + col];
        Ws[j] = w;
    }
    __syncthreads();

    int lane = threadIdx.x & 31;
    int tile = blockIdx.x * 8 + (threadIdx.x >> 5);
    int ntiles = (nrows + 15) >> 4;
    if (tile >= ntiles) return;

    int hh = lane >> 4;       // half-wave: K parity group
    int c  = lane & 15;       // column-in-tile / row-in-tile

    int arow = tile * 16 + c;
    if (arow >= nrows) arow = nrows - 1;   // clamped; garbage rows masked at store
    const float* xr = X + (long long)arow * 128;
    __builtin_prefetch(xr, 0, 1);

    v8f a0 = {0}, a1 = {0}, a2 = {0}, a3 = {0};
    for (int k = 0; k < 128; k += 4) {
        int kk = k + 2 * hh;                       // v0 holds K=2h, v1 holds K=2h+1
        v2f a = *(const v2f*)(xr + kk);
        const v2f* wp = Ws + (kk >> 1) * 64;       // K-pair row
        v2f b0 = wp[c];
        v2f b1 = wp[16 + c];
        v2f b2 = wp[32 + c];
        v2f b3 = wp[48 + c];
        a0 = __builtin_amdgcn_wmma_f32_16x16x4_f32(false, a, false, b0, (short)0, a0, false, false);
        a1 = __builtin_amdgcn_wmma_f32_16x16x4_f32(false, a, false, b1, (short)0, a1, false, false);
        a2 = __builtin_amdgcn_wmma_f32_16x16x4_f32(false, a, false, b2, (short)0, a2, false, false);
        a3 = __builtin_amdgcn_wmma_f32_16x16x4_f32(false, a, false, b3, (short)0, a3, false, false);
    }
    float* out = H + (long long)tile * 16 * 64;
    for (int r = 0; r < 8; r++) {
        int rr = r + 8 * hh;                       // C/D: vgpr r, half-wave -> M=r / r+8
        if (tile * 16 + rr >= nrows) continue;
        out[rr * 64 + c]      = a0[r];
        out[rr * 64 + 16 + c] = a1[r];
        out[rr * 64 + 32 + c] = a2[r];
        out[rr * 64 + 48 + c] = a3[r];
    }
}

// ---------------------------------------------------------------------------
// GEMM2: Hout[N,64] = BN(Hin)[N,64] @ W[64,64], BN affine fused on A load
// stats[128..191]=scale, stats[192..255]=shift; W staged as K-pairs in LDS
// ---------------------------------------------------------------------------
__global__ void __launch_bounds__(256)
gemm2_wmma(const float* __restrict__ Hin, const float* __restrict__ W,
           const float* __restrict__ stats, float* __restrict__ Hout, int nrows) {
    __shared__ v2f Ws[32 * 64];          // [kpair][col]
    __shared__ float Sc[64];
    __shared__ float Sh[64];
    for (int j = threadIdx.x; j < 32 * 64; j += 256) {
        int kp = j >> 6;
        int col = j & 63;
        v2f w;
        w.x = W[(2 * kp) * 64 + col];
        w.y = W[(2 * kp + 1) * 64 + col];
        Ws[j] = w;
    }
    if (threadIdx.x < 64) {
        Sc[threadIdx.x] = stats[128 + threadIdx.x];
        Sh[threadIdx.x] = stats[192 + threadIdx.x];
    }
    __syncthreads();

    int lane = threadIdx.x & 31;
    int tile = blockIdx.x * 8 + (threadIdx.x >> 5);
    int ntiles = (nrows + 15) >> 4;
    if (tile >= ntiles) return;

    int hh = lane >> 4;
    int c  = lane & 15;

    int arow = tile * 16 + c;
    if (arow >= nrows) arow = nrows - 1;
    const float* hr = Hin + (long long)arow * 64;
    __builtin_prefetch(hr, 0, 1);

    v8f a0 = {0}, a1 = {0}, a2 = {0}, a3 = {0};
    for (int k = 0; k < 64; k += 4) {
        int kk = k + 2 * hh;
        v2f ar = *(const v2f*)(hr + kk);
        v2f a;
        a.x = ar.x * Sc[kk]     + Sh[kk];
        a.y = ar.y * Sc[kk + 1] + Sh[kk + 1];
        const v2f* wp = Ws + (kk >> 1) * 64;
        v2f b0 = wp[c];
        v2f b1 = wp[16 + c];
        v2f b2 = wp[32 + c];
        v2f b3 = wp[48 + c];
        a0 = __builtin_amdgcn_wmma_f32_16x16x4_f32(false, a, false, b0, (short)0, a0, false, false);
        a1 = __builtin_amdgcn_wmma_f32_16x16x4_f32(false, a, false, b1, (short)0, a1, false, false);
        a2 = __builtin_amdgcn_wmma_f32_16x16x4_f32(false, a, false, b2, (short)0, a2, false, false);
        a3 = __builtin_amdgcn_wmma_f32_16x16x4_f32(false, a, false, b3, (short)0, a3, false, false);
    }
    float* out = Hout + (long long)tile * 16 * 64;
    for (int r = 0; r < 8; r++) {
        int rr = r + 8 * hh;
        if (tile * 16 + rr >= nrows) continue;
        out[rr * 64 + c]      = a0[r];
        out[rr * 64 + 16 + c] = a1[r];
        out[rr * 64 + 32 + c] = a2[r];
        out[rr * 64 + 48 + c] = a3[r];
    }
}

// ---------------------------------------------------------------------------
// per-node attention logits: alpha_s[i] = h[i,:]·as ; alpha_d[i] = h[i,:]·ad
// one wave32 per node, shuffle reduction
// ---------------------------------------------------------------------------
__global__ void __launch_bounds__(256)
node_alpha(const float* __restrict__ H, const float* __restrict__ as,
           const float* __restrict__ ad, float* __restrict__ alps,
           float* __restrict__ alpd, int N) {
    int lane = threadIdx.x & 31;
    int node = blockIdx.x * 8 + (threadIdx.x >> 5);
    if (node >= N) return;
    const float* hr = H + (long long)node * 64;
    float h0 = hr[lane], h1 = hr[lane + 32];
    float vs = h0 * as[lane] + h1 * as[lane + 32];
    float vd = h0 * ad[lane] + h1 * ad[lane + 32];
    for (int o = 16; o > 0; o >>= 1) {
        vs += __shfl_down(vs, o, 32);
        vd += __shfl_down(vd, o, 32);
    }
    if (lane == 0) { alps[node] = vs; alpd[node] = vd; }
}

// ---------------------------------------------------------------------------
// edge pass 1: e = leaky_relu(as[src]+ad[dst]); segment max via ordered atomics
// edges [0,E) from edge_index, [E,Etot) are self loops
// ---------------------------------------------------------------------------
__global__ void __launch_bounds__(256)
edge_pass1(const int* __restrict__ ei, const float* __restrict__ alps,
           const float* __restrict__ alpd, float* __restrict__ ebuf,
           unsigned* __restrict__ emax, int E, int Etot) {
    int e = blockIdx.x * 256 + threadIdx.x;
    if (e >= Etot) return;
    int s, d;
    if (e < E) { s = ei[e]; d = ei[E + e]; } else { s = d = e - E; }
    float v = alps[s] + alpd[d];
    v = (v >= 0.0f) ? v : NEG_SLOPE * v;
    ebuf[e] = v;
    atomicMax(&emax[d], f2ord(v));
}

// edge pass 2: w = exp(e - emax[dst]); denom[dst] += w
__global__ void __launch_bounds__(256)
edge_pass2(const int* __restrict__ ei, float* __restrict__ ebuf,
           const unsigned* __restrict__ emax, float* __restrict__ denom,
           int E, int Etot) {
    int e = blockIdx.x * 256 + threadIdx.x;
    if (e >= Etot) return;
    int d = (e < E) ? ei[E + e] : (e - E);
    float w = expf(ebuf[e] - ord2f(emax[d]));
    ebuf[e] = w;
    atomicAdd(&denom[d], w);
}

// edge pass 2b: alpha = w / denom[dst]
__global__ void __launch_bounds__(256)
edge_pass2b(const int* __restrict__ ei, float* __restrict__ ebuf,
            const float* __restrict__ denom, int E, int Etot) {
    int e = blockIdx.x * 256 + threadIdx.x;
    if (e >= Etot) return;
    int d = (e < E) ? ei[E + e] : (e - E);
    ebuf[e] = ebuf[e] / denom[d];   // denom > 0 guaranteed by self loop
}

// edge pass 3: agg[dst,c] += h[src,c] * alpha[e]   (thread per (edge, channel))
__global__ void __launch_bounds__(256)
edge_pass3(const int* __restrict__ ei, const float* __restrict__ ebuf,
           const float* __restrict__ H, float* __restrict__ agg,
           int E, long long total) {
    long long idx = (long long)blockIdx.x * 256 + threadIdx.x;
    if (idx >= total) return;
    int e = (int)(idx >> 6);
    int c = (int)(idx & 63);
    int s, d;
    if (e < E) { s = ei[e]; d = ei[E + e]; } else { s = d = e - E; }
    atomicAdd(&agg[(long long)d * 64 + c], H[(long long)s * 64 + c] * ebuf[e]);
}

// ---------------------------------------------------------------------------
// bias + relu in place, accumulate BN sums (stats[0:64]=sum, [64:128]=sumsq)
// block = 64 threads (one channel each), 512 rows per block
// ---------------------------------------------------------------------------
#define STAT_ROWS 512
__global__ void __launch_bounds__(64)
biasrelu_stats(float* __restrict__ agg, const float* __restrict__ bias,
               float* __restrict__ stats, int N) {
    int c = threadIdx.x;
    long long r0 = (long long)blockIdx.x * STAT_ROWS;
    float bc = bias[c];
    float s = 0.0f, s2 = 0.0f;
    for (int r = 0; r < STAT_ROWS; r++) {
        long long row = r0 + r;
        if (row >= N) break;
        float v = agg[row * 64 + c] + bc;
        v = (v > 0.0f) ? v : 0.0f;
        agg[row * 64 + c] = v;
        s += v; s2 += v * v;
    }
    atomicAdd(&stats[c], s);
    atomicAdd(&stats[64 + c], s2);
}

// fold BN into per-channel scale/shift: stats[128..]=g*rsqrt(v+eps), [192..]=b-m*scale
__global__ void __launch_bounds__(64)
stats_finalize(float* __restrict__ stats, const float* __restrict__ g,
               const float* __restrict__ b, int N) {
    int i = threadIdx.x;
    float inv = 1.0f / (float)N;
    float m = stats[i] * inv;
    float var = stats[64 + i] * inv - m * m;
    float sc = rsqrtf(var + BN_EPS_C) * g[i];
    stats[128 + i] = sc;
    stats[192 + i] = b[i] - m * sc;
}

// global mean pool (BN2 applied on the fly): pool[g,c] += bn(h[n,c]); cnt[g]+=1
__global__ void __launch_bounds__(256)
pool_kernel(const float* __restrict__ h, const float* __restrict__ stats,
            const int* __restrict__ batch, float* __restrict__ pool,
            float* __restrict__ cnt, int N) {
    long long idx = (long long)blockIdx.x * 256 + threadIdx.x;
    if (idx >= (long long)N * 64) return;
    int node = (int)(idx >> 6);
    int c = (int)(idx & 63);
    float v = h[idx] * stats[128 + c] + stats[192 + c];
    int g = batch[node];
    atomicAdd(&pool[(long long)g * 64 + c], v);
    if (c == 0) atomicAdd(&cnt[g], 1.0f);
}

// MLP head: 64 -> 64 -> 16 -> 8 -> 10, one block per graph
__global__ void __launch_bounds__(64)
mlp_head(const float* __restrict__ pool, const float* __restrict__ cnt,
         const float* __restrict__ m1w, const float* __restrict__ m1b,
         const float* __restrict__ m2w, const float* __restrict__ m2b,
         const float* __restrict__ m3w, const float* __restrict__ m3b,
         const float* __restrict__ m4w, const float* __restrict__ m4b,
         float* __restrict__ out) {
    __shared__ float p[64], z1[64], z2[16], z3[8];
    int g = blockIdx.x, t = threadIdx.x;
    float c = cnt[g]; c = (c > 1.0f) ? c : 1.0f;
    p[t] = pool[(long long)g * 64 + t] / c;
    __syncthreads();
    { float s = m1b[t]; for (int k = 0; k < 64; k++) s += p[k] * m1w[k * 64 + t];
      z1[t] = (s > 0.0f) ? s : 0.0f; }
    __syncthreads();
    if (t < 16) { float s = m2b[t]; for (int k = 0; k < 64; k++) s += z1[k] * m2w[k * 16 + t];
                  z2[t] = (s > 0.0f) ? s : 0.0f; }
    __syncthreads();
    if (t < 8) { float s = m3b[t]; for (int k = 0; k < 16; k++) s += z2[k] * m3w[k * 8 + t];
                 z3[t] = (s > 0.0f) ? s : 0.0f; }
    __syncthreads();
    if (t < 10) { float s = m4b[t]; for (int k = 0; k < 8; k++) s += z3[k] * m4w[k * 10 + t];
                  out[(long long)g * 10 + t] = s; }
}

// ---------------------------------------------------------------------------
extern "C" void kernel_launch(void* const* d_in, const int* in_sizes, int n_in,
                              void* d_out, int out_size, void* d_ws, size_t ws_size,
                              hipStream_t stream) {
    const float* x     = (const float*)d_in[0];
    const int*   ei    = (const int*)  d_in[1];
    const int*   batch = (const int*)  d_in[2];
    const float* W1  = (const float*)d_in[3];
    const float* as1 = (const float*)d_in[4];
    const float* ad1 = (const float*)d_in[5];
    const float* b1  = (const float*)d_in[6];
    const float* W2  = (const float*)d_in[7];
    const float* as2 = (const float*)d_in[8];
    const float* ad2 = (const float*)d_in[9];
    const float* b2  = (const float*)d_in[10];
    const float* bng = (const float*)d_in[11];
    const float* bnb = (const float*)d_in[12];
    const float* m1w = (const float*)d_in[13];
    const float* m1b = (const float*)d_in[14];
    const float* m2w = (const float*)d_in[15];
    const float* m2b = (const float*)d_in[16];
    const float* m3w = (const float*)d_in[17];
    const float* m3b = (const float*)d_in[18];
    const float* m4w = (const float*)d_in[19];
    const float* m4b = (const float*)d_in[20];
    float* out = (float*)d_out;

    const int N = in_sizes[0] / 128;
    const int E = in_sizes[1] / 2;
    const int G = out_size / 10;
    const int Etot = E + N;

    // workspace layout
    char* ws = (char*)d_ws;
    auto alloc = [&](size_t bytes) -> void* {
        void* p = (void*)ws;
        ws += (bytes + 255) & ~(size_t)255;
        return p;
    };
    float*    H1    = (float*)   alloc((size_t)N * 64 * 4);  // gemm out (h1, then h2)
    float*    AGG   = (float*)   alloc((size_t)N * 64 * 4);  // aggregation / bn input
    float*    ALPS  = (float*)   alloc((size_t)N * 4);
    float*    ALPD  = (float*)   alloc((size_t)N * 4);
    float*    EBUF  = (float*)   alloc((size_t)Etot * 4);
    unsigned* EMAX  = (unsigned*)alloc((size_t)N * 4);
    float*    DENOM = (float*)   alloc((size_t)N * 4);
    float*    STATS = (float*)   alloc(256 * 4);             // sum|sumsq|scale|shift
    float*    POOL  = (float*)   alloc((size_t)G * 64 * 4);
    float*    CNT   = (float*)   alloc((size_t)G * 4);

    const int gInit  = (int)(((long long)N * 64 + 255) / 256);
    const int gGemm  = ((N + 15) / 16 + 7) / 8;
    const int gNode  = (N + 7) / 8;
    const int gEdge  = (Etot + 255) / 256;
    const long long agg_total = (long long)Etot * 64;
    const int gAgg   = (int)((agg_total + 255) / 256);
    const int gStat  = (N + STAT_ROWS - 1) / STAT_ROWS;

    // ---------------- layer 1 ----------------
    init_layer<<<gInit, 256, 0, stream>>>(AGG, EMAX, DENOM, STATS, POOL, CNT, N, G);
    gemm1_wmma<<<gGemm, 256, 0, stream>>>(x, W1, H1, N);
    node_alpha<<<gNode, 256, 0, stream>>>(H1, as1, ad1, ALPS, ALPD, N);
    edge_pass1<<<gEdge, 256, 0, stream>>>(ei, ALPS, ALPD, EBUF, EMAX, E, Etot);
    edge_pass2<<<gEdge, 256, 0, stream>>>(ei, EBUF, EMAX, DENOM, E, Etot);
    edge_pass2b<<<gEdge, 256, 0, stream>>>(ei, EBUF, DENOM, E, Etot);
    edge_pass3<<<gAgg, 256, 0, stream>>>(ei, EBUF, H1, AGG, E, agg_total);
    biasrelu_stats<<<gStat, 64, 0, stream>>>(AGG, b1, STATS, N);
    stats_finalize<<<1, 64, 0, stream>>>(STATS, bng, bnb, N);

    // ---------------- layer 2 ----------------
    gemm2_wmma<<<gGemm, 256, 0, stream>>>(AGG, W2, STATS, H1, N);  // reads BN1 stats
    init_layer<<<gInit, 256, 0, stream>>>(AGG, EMAX, DENOM, STATS, POOL, CNT, N, G);
    node_alpha<<<gNode, 256, 0, stream>>>(H1, as2, ad2, ALPS, ALPD, N);
    edge_pass1<<<gEdge, 256, 0, stream>>>(ei, ALPS, ALPD, EBUF, EMAX, E, Etot);
    edge_pass2<<<gEdge, 256, 0, stream>>>(ei, EBUF, EMAX, DENOM, E, Etot);
    edge_pass2b<<<gEdge, 256, 0, stream>>>(ei, EBUF, DENOM, E, Etot);
    edge_pass3<<<gAgg, 256, 0, stream>>>(ei, EBUF, H1, AGG, E, agg_total);
    biasrelu_stats<<<gStat, 64, 0, stream>>>(AGG, b2, STATS, N);
    stats_finalize<<<1, 64, 0, stream>>>(STATS, bng, bnb, N);

    // ---------------- pool + head ----------------
    pool_kernel<<<gInit, 256, 0, stream>>>(AGG, STATS, batch, POOL, CNT, N);
    mlp_head<<<G, 64, 0, stream>>>(POOL, CNT, m1w, m1b, m2w, m2b, m3w, m3b,
                                   m4w, m4b, out);
}